// ResGATsNoResidual_59906203845001
// MI455X (gfx1250) — compile-verified
//
#include <hip/hip_runtime.h>
#include <cstdint>
#include <cstddef>

typedef __attribute__((ext_vector_type(2))) float v2f;
typedef __attribute__((ext_vector_type(8))) float v8f;

#define LEAKY_SLOPE 0.2f
#define GN_EPS 1e-5f

// ---- order-preserving float <-> uint for atomicMax-based segment max ----
__device__ __forceinline__ unsigned f32_ord(float f) {
    unsigned u = __float_as_uint(f);
    return (u & 0x80000000u) ? ~u : (u | 0x80000000u);
}
__device__ __forceinline__ float ord_f32(unsigned u) {
    unsigned b = (u & 0x80000000u) ? (u & 0x7fffffffu) : ~u;
    return __uint_as_float(b);
}

__device__ __forceinline__ v8f bcast8(float v) {
    v8f r = { v, v, v, v, v, v, v, v };
    return r;
}
__device__ __forceinline__ v8f wmma_f32(v2f a, v2f b, v8f c) {
    return __builtin_amdgcn_wmma_f32_16x16x4_f32(
        /*neg_a=*/false, a, /*neg_b=*/false, b,
        /*c_mod=*/(short)0, c, /*reuse_a=*/false, /*reuse_b=*/false);
}

// =====================================================================
// Fused dual GEMM: XL = A@Wl + bl, XR = A@Wr + br  (f32 WMMA 16x16x4)
// A is shared between both products -> fetched once.
// One wave computes 16 rows x 32 cols of BOTH XL and XR (4 WMMA tiles,
// 4 independent accumulator chains). 8 waves / block.
// Strength-reduced: 4 running W row pointers advance by 4*Dout floats
// per K-step; loads use immediate offsets only.
// Requires: N%16==0, K%4==0, Dout%32==0, total wave-tiles %8==0 (true here).
// A layout (16x4 f32):  lane m = lane%16; vgpr{0,1} hold K = 2*(lane/16)+{0,1}
// B layout (4x16 f32):  lane n = lane%16; vgpr{0,1} hold K = 2*(lane/16)+{0,1}
// C/D layout: row = vgpr + 8*(lane/16), col = lane%16
// =====================================================================
__global__ __launch_bounds__(256) void k_gemm2_bias_wmma(
    const float* __restrict__ A,
    const float* __restrict__ Wl, const float* __restrict__ bl,
    const float* __restrict__ Wr, const float* __restrict__ br,
    float* __restrict__ XL, float* __restrict__ XR,
    int K, int Dout)
{
    const int lane    = threadIdx.x & 31;
    const int wave    = threadIdx.x >> 5;
    const int tiles_n = Dout >> 5;                 // 32-column groups
    const int t       = blockIdx.x * 8 + wave;
    const int tn      = t % tiles_n;
    const int tm      = t / tiles_n;
    const int l16     = lane & 15;
    const int half    = lane >> 4;                 // 0 or 1
    const int kh      = half * 2;
    const int col0    = tn * 32 + l16;             // tile cols: col0, col0+16
    const int arow    = tm * 16 + l16;

    v8f cl0 = bcast8(bl[col0]);
    v8f cl1 = bcast8(bl[col0 + 16]);
    v8f cr0 = bcast8(br[col0]);
    v8f cr1 = bcast8(br[col0 + 16]);

    const size_t DS = (size_t)Dout;                // row stride (floats)
    const float* __restrict__ pA   = A  + (size_t)arow * K + kh;
    const float* __restrict__ pwl0 = Wl + (size_t)kh * DS + col0;   // row k+kh
    const float* __restrict__ pwl1 = pwl0 + DS;                     // row k+kh+1
    const float* __restrict__ pwr0 = Wr + (size_t)kh * DS + col0;
    const float* __restrict__ pwr1 = pwr0 + DS;
    const size_t step = 4 * DS;                    // K-step advance (floats)

    for (int k = 0; k < K; k += 4) {
        v2f a = *(const v2f*)pA;

        // WGP-scope prefetch of next K-step's streaming B rows
        __builtin_prefetch(pwl0 + step, 0, 3);
        __builtin_prefetch(pwl1 + step, 0, 3);
        __builtin_prefetch(pwr0 + step, 0, 3);
        __builtin_prefetch(pwr1 + step, 0, 3);

        v2f b;
        b.x = pwl0[0];  b.y = pwl1[0];  cl0 = wmma_f32(a, b, cl0);
        b.x = pwl0[16]; b.y = pwl1[16]; cl1 = wmma_f32(a, b, cl1);
        b.x = pwr0[0];  b.y = pwr1[0];  cr0 = wmma_f32(a, b, cr0);
        b.x = pwr0[16]; b.y = pwr1[16]; cr1 = wmma_f32(a, b, cr1);

        pA   += 4;
        pwl0 += step; pwl1 += step;
        pwr0 += step; pwr1 += step;
    }

    const size_t ob = (size_t)(tm * 16 + half * 8) * DS + col0;
#pragma unroll
    for (int v = 0; v < 8; ++v) {
        XL[ob + (size_t)v * DS]      = cl0[v];
        XL[ob + (size_t)v * DS + 16] = cl1[v];
        XR[ob + (size_t)v * DS]      = cr0[v];
        XR[ob + (size_t)v * DS + 16] = cr1[v];
    }
}

// ---------------------------------------------------------------------
// Fill helpers
// ---------------------------------------------------------------------
__global__ void k_fill_f32(float* p, float v, int n) {
    int i = blockIdx.x * blockDim.x + threadIdx.x;
    if (i < n) p[i] = v;
}
__global__ void k_fill_u32(unsigned* p, unsigned v, int n) {
    int i = blockIdx.x * blockDim.x + threadIdx.x;
    if (i < n) p[i] = v;
}
__global__ void k_fill_bias(float* p, const float* __restrict__ b, int n, int D) {
    int i = blockIdx.x * blockDim.x + threadIdx.x;
    if (i < n) p[i] = b[i % D];
}

// ---------------------------------------------------------------------
// Edge scoring: a[e][h] = sum_c att[h][c]*leakyrelu(xl[src][h*C+c]+xr[dst][h*C+c])
// also amax[dst][h] = max(...). One wave per edge (self-loops appended).
// ---------------------------------------------------------------------
__global__ __launch_bounds__(256) void k_edge_scores(
    const float* __restrict__ xl, const float* __restrict__ xr,
    const int* __restrict__ src, const int* __restrict__ dst,
    const float* __restrict__ att, float* __restrict__ a,
    unsigned* __restrict__ amax, int E, int N, int H, int C)
{
    const int e    = blockIdx.x * 8 + (threadIdx.x >> 5);
    const int lane = threadIdx.x & 31;
    const int Et   = E + N;
    if (e >= Et) return;
    const int s = (e < E) ? src[e] : (e - E);
    const int d = (e < E) ? dst[e] : (e - E);
    const int D = H * C;
    const float* __restrict__ pl = xl + (size_t)s * D;
    const float* __restrict__ pr = xr + (size_t)d * D;
    for (int h = 0; h < H; ++h) {
        const int base = h * C;
        float acc = 0.f;
        for (int c0 = lane; c0 < C; c0 += 32) {
            float m  = pl[base + c0] + pr[base + c0];
            float lr = (m > 0.f) ? m : LEAKY_SLOPE * m;
            acc += lr * att[base + c0];
        }
#pragma unroll
        for (int off = 16; off > 0; off >>= 1)
            acc += __shfl_xor(acc, off, 32);
        if (lane == 0) {
            a[(size_t)e * H + h] = acc;
            atomicMax(&amax[(size_t)d * H + h], f32_ord(acc));
        }
    }
}

// ---------------------------------------------------------------------
// Softmax numerator: a[e][h] = exp(a - amax[dst][h]); denom[dst][h] += it
// ---------------------------------------------------------------------
__global__ void k_edge_expsum(
    float* __restrict__ a, const unsigned* __restrict__ amax,
    const int* __restrict__ dst, float* __restrict__ denom,
    int E, int N, int H)
{
    const int idx = blockIdx.x * blockDim.x + threadIdx.x;
    const int Et  = E + N;
    if (idx >= Et * H) return;
    const int e = idx / H, h = idx - e * H;
    const int d = (e < E) ? dst[e] : (e - E);
    const float mx = ord_f32(amax[(size_t)d * H + h]);
    const float w  = expf(a[idx] - mx);
    a[idx] = w;
    atomicAdd(&denom[(size_t)d * H + h], w);
}

// ---------------------------------------------------------------------
// Weighted aggregation: out[dst] += (a[e][h]/(denom+1e-16)) * xl[src]
// out pre-initialized with per-channel bias. One wave per edge.
// ---------------------------------------------------------------------
__global__ __launch_bounds__(256) void k_edge_aggregate(
    const float* __restrict__ xl, const float* __restrict__ a,
    const float* __restrict__ denom,
    const int* __restrict__ src, const int* __restrict__ dst,
    float* __restrict__ out, int E, int N, int H, int C)
{
    const int e    = blockIdx.x * 8 + (threadIdx.x >> 5);
    const int lane = threadIdx.x & 31;
    const int Et   = E + N;
    if (e >= Et) return;
    const int s = (e < E) ? src[e] : (e - E);
    const int d = (e < E) ? dst[e] : (e - E);
    const int D = H * C;
    const float* __restrict__ pl = xl + (size_t)s * D;
    float* __restrict__ po = out + (size_t)d * D;
    for (int h = 0; h < H; ++h) {
        const float w = a[(size_t)e * H + h] /
                        (denom[(size_t)d * H + h] + 1e-16f);
        const int base = h * C;
        for (int c0 = lane; c0 < C; c0 += 32)
            atomicAdd(&po[base + c0], w * pl[base + c0]);
    }
}

// ---------------------------------------------------------------------
// GraphNorm pieces (batch is per-node graph id, G graphs)
// ---------------------------------------------------------------------
__global__ void k_node_count(const int* __restrict__ batch, float* cnt, int N) {
    int i = blockIdx.x * blockDim.x + threadIdx.x;
    if (i < N) atomicAdd(&cnt[batch[i]], 1.f);
}
__global__ void k_seg_sum(const float* __restrict__ h, const int* __restrict__ batch,
                          float* __restrict__ sum, int N, int D)
{
    int idx = blockIdx.x * blockDim.x + threadIdx.x;
    if (idx >= N * D) return;
    int n = idx / D, d = idx - n * D;
    atomicAdd(&sum[(size_t)batch[n] * D + d], h[idx]);
}
__global__ void k_gn_center(float* __restrict__ h, const int* __restrict__ batch,
                            const float* __restrict__ sum, const float* __restrict__ cnt,
                            const float* __restrict__ ms, float* __restrict__ var,
                            int N, int D)
{
    int idx = blockIdx.x * blockDim.x + threadIdx.x;
    if (idx >= N * D) return;
    int n = idx / D, d = idx - n * D;
    int g = batch[n];
    float c = fmaxf(cnt[g], 1.f);
    float mean = sum[(size_t)g * D + d] / c;
    float o = h[idx] - mean * ms[d];
    h[idx] = o;
    atomicAdd(&var[(size_t)g * D + d], o * o);
}
__global__ void k_gn_final(float* __restrict__ h, const int* __restrict__ batch,
                           const float* __restrict__ var, const float* __restrict__ cnt,
                           const float* __restrict__ w, const float* __restrict__ b,
                           int N, int D)
{
    int idx = blockIdx.x * blockDim.x + threadIdx.x;
    if (idx >= N * D) return;
    int n = idx / D, d = idx - n * D;
    int g = batch[n];
    float c = fmaxf(cnt[g], 1.f);
    float v = var[(size_t)g * D + d] / c;
    float y = w[d] * h[idx] / sqrtf(v + GN_EPS) + b[d];
    h[idx] = (y > 0.f) ? y : expm1f(y);   // ELU
}

// ---------------------------------------------------------------------
// Head: pooled[g] = pool[g]/cnt[g]; z = elu(pooled@W1+b1); logits = z@W2+b2
// out[0:64] = logits, out[64:128] = log_softmax(logits). One block per graph.
// ---------------------------------------------------------------------
__global__ __launch_bounds__(256) void k_head_mlp(
    const float* __restrict__ pool, const float* __restrict__ cnt,
    const float* __restrict__ w1, const float* __restrict__ b1,
    const float* __restrict__ w2, const float* __restrict__ b2,
    float* __restrict__ out)
{
    __shared__ float sp[256];
    __shared__ float sz[512];
    __shared__ float r0[256], r1[256];
    const int g = blockIdx.x, t = threadIdx.x;
    const float c = fmaxf(cnt[g], 1.f);
    sp[t] = pool[(size_t)g * 256 + t] / c;
    __syncthreads();
    for (int j = t; j < 512; j += 256) {
        float acc = b1[j];
        for (int d = 0; d < 256; ++d)
            acc += sp[d] * w1[(size_t)d * 512 + j];
        sz[j] = (acc > 0.f) ? acc : expm1f(acc);
    }
    __syncthreads();
    float a0 = 0.f, a1 = 0.f;
    for (int j = t; j < 512; j += 256) {
        a0 += sz[j] * w2[j * 2 + 0];
        a1 += sz[j] * w2[j * 2 + 1];
    }
    r0[t] = a0; r1[t] = a1;
    __syncthreads();
    for (int s = 128; s > 0; s >>= 1) {
        if (t < s) { r0[t] += r0[t + s]; r1[t] += r1[t + s]; }
        __syncthreads();
    }
    if (t == 0) {
        float l0 = r0[0] + b2[0], l1 = r1[0] + b2[1];
        out[g * 2 + 0] = l0;
        out[g * 2 + 1] = l1;
        float mx  = fmaxf(l0, l1);
        float lse = mx + logf(expf(l0 - mx) + expf(l1 - mx));
        out[64 + g * 2 + 0] = l0 - lse;
        out[64 + g * 2 + 1] = l1 - lse;
    }
}

// =====================================================================
// Host orchestration
// =====================================================================
static inline int cdiv(int a, int b) { return (a + b - 1) / b; }

extern "C" void kernel_launch(void* const* d_in, const int* in_sizes, int n_in,
                              void* d_out, int out_size, void* d_ws, size_t ws_size,
                              hipStream_t stream)
{
    const int DIN = 256, G = 32;
    const int N = in_sizes[0] / DIN;        // 20000
    const int E = in_sizes[1] / 2;          // 320000
    const int Et = E + N;                   // with self-loops
    const int MAXD = 1024;

    const float* x    = (const float*)d_in[0];
    const int*   esrc = (const int*)d_in[1];
    const int*   edst = esrc + E;
    const int*   batch = (const int*)d_in[2];
    const float* fc1w = (const float*)d_in[30];
    const float* fc1b = (const float*)d_in[31];
    const float* fc2w = (const float*)d_in[32];
    const float* fc2b = (const float*)d_in[33];
    float* out = (float*)d_out;

    // --- workspace carve-up (256B aligned) ---
    size_t off = 0;
    auto carve = [&](size_t bytes) -> void* {
        void* p = (char*)d_ws + off;
        off += (bytes + 255) & ~(size_t)255;
        return p;
    };
    float*    hA    = (float*)carve((size_t)N * MAXD * 4);
    float*    hB    = (float*)carve((size_t)N * MAXD * 4);
    float*    xl    = (float*)carve((size_t)N * MAXD * 4);
    float*    xr    = (float*)carve((size_t)N * MAXD * 4);
    float*    aE    = (float*)carve((size_t)Et * 8 * 4);
    unsigned* amax  = (unsigned*)carve((size_t)N * 8 * 4);
    float*    denom = (float*)carve((size_t)N * 8 * 4);
    float*    cnt   = (float*)carve((size_t)G * 4);
    float*    ssum  = (float*)carve((size_t)G * MAXD * 4);
    float*    svar  = (float*)carve((size_t)G * MAXD * 4);
    float*    pool  = (float*)carve((size_t)G * 256 * 4);
    (void)ws_size; (void)n_in; (void)out_size;

    const int dinA[3] = {256, 1024, 512};
    const int Harr[3] = {8, 2, 4};
    const int Carr[3] = {128, 256, 64};

    const float* hin = x;
    float* hout = hA;

    for (int l = 0; l < 3; ++l) {
        const int K = dinA[l], H = Harr[l], C = Carr[l], D = H * C;
        const int base = 3 + 9 * l;
        const float* Wl  = (const float*)d_in[base + 0];
        const float* bl  = (const float*)d_in[base + 1];
        const float* Wr  = (const float*)d_in[base + 2];
        const float* br  = (const float*)d_in[base + 3];
        const float* att = (const float*)d_in[base + 4];
        const float* bi  = (const float*)d_in[base + 5];
        const float* gw  = (const float*)d_in[base + 6];
        const float* gb  = (const float*)d_in[base + 7];
        const float* gms = (const float*)d_in[base + 8];

        // 1) fused dual GEMM via f32 WMMA: one wave = 16x32 of XL AND XR
        //    wave-tiles = (N/16)*(D/32); always divisible by 8 here.
        const int wtiles = (N / 16) * (D / 32);
        k_gemm2_bias_wmma<<<wtiles / 8, 256, 0, stream>>>(
            hin, Wl, bl, Wr, br, xl, xr, K, D);

        // 2) attention scores + segment max
        k_fill_u32<<<cdiv(N * H, 256), 256, 0, stream>>>(amax, 0u, N * H);
        k_fill_f32<<<cdiv(N * H, 256), 256, 0, stream>>>(denom, 0.f, N * H);
        k_edge_scores<<<cdiv(Et, 8), 256, 0, stream>>>(
            xl, xr, esrc, edst, att, aE, amax, E, N, H, C);

        // 3) exp + denominator
        k_edge_expsum<<<cdiv(Et * H, 256), 256, 0, stream>>>(
            aE, amax, edst, denom, E, N, H);

        // 4) weighted scatter-add (out pre-seeded with bias)
        k_fill_bias<<<cdiv(N * D, 256), 256, 0, stream>>>(hout, bi, N * D, D);
        k_edge_aggregate<<<cdiv(Et, 8), 256, 0, stream>>>(
            xl, aE, denom, esrc, edst, hout, E, N, H, C);

        // 5) GraphNorm + ELU (in place on hout)
        k_fill_f32<<<cdiv(G * D, 256), 256, 0, stream>>>(ssum, 0.f, G * D);
        k_fill_f32<<<cdiv(G * D, 256), 256, 0, stream>>>(svar, 0.f, G * D);
        k_fill_f32<<<1, G, 0, stream>>>(cnt, 0.f, G);
        k_node_count<<<cdiv(N, 256), 256, 0, stream>>>(batch, cnt, N);
        k_seg_sum<<<cdiv(N * D, 256), 256, 0, stream>>>(hout, batch, ssum, N, D);
        k_gn_center<<<cdiv(N * D, 256), 256, 0, stream>>>(
            hout, batch, ssum, cnt, gms, svar, N, D);
        k_gn_final<<<cdiv(N * D, 256), 256, 0, stream>>>(
            hout, batch, svar, cnt, gw, gb, N, D);

        // ping-pong
        hin  = hout;
        hout = (hout == hA) ? hB : hA;
    }

    // mean pool (final features live in `hin`, D=256) + MLP head
    k_fill_f32<<<cdiv(G * 256, 256), 256, 0, stream>>>(pool, 0.f, G * 256);
    k_fill_f32<<<1, G, 0, stream>>>(cnt, 0.f, G);
    k_node_count<<<cdiv(N, 256), 256, 0, stream>>>(batch, cnt, N);
    k_seg_sum<<<cdiv(N * 256, 256), 256, 0, stream>>>(hin, batch, pool, N, 256);
    k_head_mlp<<<G, 256, 0, stream>>>(pool, cnt, fc1w, fc1b, fc2w, fc2b, out);
}